// GatedSelfAttention_42666205118982
// MI455X (gfx1250) — compile-verified
//
#include <hip/hip_runtime.h>
#include <math.h>

// ---------------------------------------------------------------------------
// GatedSelfAttention decode step for MI455X (gfx1250, wave32).
//
// Roofline: dominated by 302MB (QKV) + 134MB (o_proj) fp32 weight streaming
// -> GEMV kernels optimized for coalesced float4 HBM streams.
// Attention (4 KV heads, 8 q-rows each, HEAD_DIM=256, seq<=2048) is the real
// matrix work -> V_WMMA_F32_16X16X4_F32 (exact fp32, matches reference).
// ---------------------------------------------------------------------------

#define NUM_HEADS    32
#define NUM_KV_HEADS 4
#define GROUPS       8
#define HEAD_DIM     256
#define HIDDEN       4096
#define MAX_SEQ      2048
#define ROTARY_HALF  32
#define THETA        10000000.0f
#define SCALE        0.0625f   // 1/sqrt(256)

typedef __attribute__((ext_vector_type(2))) float v2f;
typedef __attribute__((ext_vector_type(8))) float v8f;

// ---------------- Kernel 1: fused QKV GEMV ---------------------------------
// 18432 output rows (16384 qg + 1024 k + 1024 v), each = dot(x[4096], W_row).
// x staged in LDS; one wave per row; lane reads contiguous float4 -> 512B/instr.
__global__ void k_gemv_qkv(const float* __restrict__ x,
                           const float* __restrict__ qw,
                           const float* __restrict__ kw,
                           const float* __restrict__ vw,
                           float* __restrict__ qg,
                           float* __restrict__ klin,
                           float* __restrict__ vlin) {
  __shared__ float xs[HIDDEN];
  const int tid = threadIdx.x;
  for (int i = tid; i < HIDDEN; i += 256) xs[i] = x[i];
  __syncthreads();

  const int lane = tid & 31;
  const int wv = tid >> 5;
  const int row = blockIdx.x * 8 + wv;          // 0..18431

  const float* wrow;
  float* dst;
  if (row < 16384)      { wrow = qw + (size_t)row * HIDDEN;           dst = qg   + row; }
  else if (row < 17408) { wrow = kw + (size_t)(row - 16384) * HIDDEN; dst = klin + (row - 16384); }
  else                  { wrow = vw + (size_t)(row - 17408) * HIDDEN; dst = vlin + (row - 17408); }

  const float4* w4 = reinterpret_cast<const float4*>(wrow);
  const float4* x4 = reinterpret_cast<const float4*>(xs);
  float acc = 0.f;
#pragma unroll 8
  for (int it = 0; it < HIDDEN / 128; ++it) {   // 32 iters of float4
    float4 wv4 = w4[it * 32 + lane];
    float4 xv4 = x4[it * 32 + lane];
    acc += wv4.x * xv4.x + wv4.y * xv4.y + wv4.z * xv4.z + wv4.w * xv4.w;
  }
  for (int off = 16; off > 0; off >>= 1) acc += __shfl_down(acc, off, 32);
  if (lane == 0) *dst = acc;
}

// ---------------- Kernel 2: per-head RMSNorm + partial RoPE ----------------
// blocks 0..31: q heads (from qg, stride 512, first 256 = q); write to q_ws
//   laid out as [g][16 rows][256] (rows 8..15 are zero padding for WMMA M=16).
// blocks 32..35: k heads; write roped k row to krot.
// blocks 36..39: zero q_ws rows 8..15 of group g.
__global__ void k_norm_rope(const float* __restrict__ qg,
                            const float* __restrict__ klin,
                            const int* __restrict__ posp,
                            const float* __restrict__ qnw,
                            const float* __restrict__ knw,
                            float* __restrict__ q_ws,
                            float* __restrict__ krot) {
  __shared__ float sh[256];
  __shared__ float vals[256];
  const int b = blockIdx.x, tid = threadIdx.x;

  if (b >= 36) {                                 // zero-pad rows 8..15
    int g = b - 36;
    float* base = q_ws + (size_t)(g * 16 + 8) * HEAD_DIM;
    for (int i = tid; i < 8 * HEAD_DIM; i += 256) base[i] = 0.f;
    return;
  }

  const bool isQ = (b < 32);
  float v = isQ ? qg[b * 512 + tid] : klin[(b - 32) * HEAD_DIM + tid];

  sh[tid] = v * v;
  __syncthreads();
  for (int off = 128; off > 0; off >>= 1) {
    if (tid < off) sh[tid] += sh[tid + off];
    __syncthreads();
  }
  const float norm = rsqrtf(sh[0] / (float)HEAD_DIM + 1e-6f);
  const float w = isQ ? qnw[tid] : knw[tid];
  const float nv = v * norm * w;
  __syncthreads();
  vals[tid] = nv;
  __syncthreads();

  float o = nv;
  if (tid < 2 * ROTARY_HALF) {
    const float pos = (float)posp[0];
    const int i = tid & (ROTARY_HALF - 1);
    const float fr = pos * powf(THETA, -(float)i / (float)ROTARY_HALF);
    float s, c;
    sincosf(fr, &s, &c);
    o = (tid < ROTARY_HALF) ? (vals[tid] * c - vals[tid + ROTARY_HALF] * s)
                            : (vals[tid] * c + vals[tid - ROTARY_HALF] * s);
  }
  if (isQ) {
    int g = b >> 3, m = b & 7;
    q_ws[(size_t)(g * 16 + m) * HEAD_DIM + tid] = o;
  } else {
    krot[(b - 32) * HEAD_DIM + tid] = o;
  }
}

// ---------------- Kernel 3: attention scores via f32 WMMA ------------------
// One wave per (kv head g, 16-position tile). D(16x16) = Q(16x256) x K^T tile,
// accumulated in 64 V_WMMA_F32_16X16X4_F32 steps. Row s==pos uses fresh krot.
__global__ void k_scores(const float* __restrict__ q_ws,
                         const float* __restrict__ krot,
                         const float* __restrict__ kcache,
                         const int* __restrict__ posp,
                         float* __restrict__ sp) {
  const int tid = threadIdx.x;
  const int lane = tid & 31;
  const int wv = tid >> 5;
  const int tile = blockIdx.x * 4 + wv;          // 0..511
  const int g = tile >> 7;
  const int s0 = (tile & 127) << 4;
  const int pos = posp[0];

  const int nm = lane & 15;                      // A: row m / B: col n
  const int koff = (lane >> 4) << 1;             // K pair select (0 or 2)

  const float* qrow = q_ws + (size_t)(g * 16 + nm) * HEAD_DIM;
  const int s = s0 + nm;
  const float* krow = (s == pos) ? (krot + g * HEAD_DIM)
                                 : (kcache + ((size_t)g * MAX_SEQ + s) * HEAD_DIM);

  v8f c = {0.f, 0.f, 0.f, 0.f, 0.f, 0.f, 0.f, 0.f};
#pragma unroll 4
  for (int k0 = 0; k0 < HEAD_DIM; k0 += 4) {
    v2f av, bv;
    av.x = qrow[k0 + koff];
    av.y = qrow[k0 + koff + 1];
    bv.x = krow[k0 + koff];
    bv.y = krow[k0 + koff + 1];
    c = __builtin_amdgcn_wmma_f32_16x16x4_f32(false, av, false, bv,
                                              (short)0, c, false, false);
  }
  if (lane < 16) {                               // rows 0..7 live in lanes 0..15
    float* dst = sp + (size_t)(g * 16) * MAX_SEQ + s0 + lane;
#pragma unroll
    for (int r = 0; r < 8; ++r) dst[(size_t)r * MAX_SEQ] = c[r] * SCALE;
  }
}

// ---------------- Kernel 4: masked softmax (in place) ----------------------
// blocks 0..31: real rows (g,m). blocks 32..63: zero padding rows 8..15.
__global__ void k_softmax(float* __restrict__ sp, const int* __restrict__ posp) {
  __shared__ float sh[256];
  const int b = blockIdx.x, tid = threadIdx.x;
  const int pos = posp[0];

  if (b >= 32) {
    int idx = b - 32;
    float* row = sp + (size_t)((idx >> 3) * 16 + 8 + (idx & 7)) * MAX_SEQ;
    for (int i = tid; i < MAX_SEQ; i += 256) row[i] = 0.f;
    return;
  }
  float* row = sp + (size_t)((b >> 3) * 16 + (b & 7)) * MAX_SEQ;

  float vals[8];
  float mx = -3.4e38f;
#pragma unroll
  for (int i = 0; i < 8; ++i) {
    int s = tid + i * 256;
    vals[i] = row[s];
    if (s <= pos) mx = fmaxf(mx, vals[i]);
  }
  sh[tid] = mx;
  __syncthreads();
  for (int off = 128; off > 0; off >>= 1) {
    if (tid < off) sh[tid] = fmaxf(sh[tid], sh[tid + off]);
    __syncthreads();
  }
  mx = sh[0];
  __syncthreads();

  float ev[8];
  float sum = 0.f;
#pragma unroll
  for (int i = 0; i < 8; ++i) {
    int s = tid + i * 256;
    ev[i] = (s <= pos) ? expf(vals[i] - mx) : 0.f;
    sum += ev[i];
  }
  sh[tid] = sum;
  __syncthreads();
  for (int off = 128; off > 0; off >>= 1) {
    if (tid < off) sh[tid] += sh[tid + off];
    __syncthreads();
  }
  const float inv = 1.f / sh[0];
#pragma unroll
  for (int i = 0; i < 8; ++i) row[tid + i * 256] = ev[i] * inv;
}

// ---------------- Kernel 5: P x V via f32 WMMA -----------------------------
// One wave per (g, 16-dim tile). D(16x16) += P(16x4) x V(4x16) over s<=pos.
__global__ void k_pv(const float* __restrict__ sp,
                     const float* __restrict__ vlin,
                     const float* __restrict__ vcache,
                     const int* __restrict__ posp,
                     float* __restrict__ attn) {
  const int tid = threadIdx.x;
  const int lane = tid & 31;
  const int wv = tid >> 5;
  const int tile = blockIdx.x * 4 + wv;          // 0..63
  const int g = tile >> 4;
  const int d0 = (tile & 15) << 4;
  const int pos = posp[0];

  const int nm = lane & 15;
  const int koff = (lane >> 4) << 1;

  const float* prow = sp + (size_t)(g * 16 + nm) * MAX_SEQ;
  const float* vbase = vcache + (size_t)g * MAX_SEQ * HEAD_DIM;
  const float* vnew = vlin + g * HEAD_DIM;

  v8f c = {0.f, 0.f, 0.f, 0.f, 0.f, 0.f, 0.f, 0.f};
  for (int s0 = 0; s0 <= pos; s0 += 4) {         // p==0 beyond pos -> safe
    v2f av, bv;
    av.x = prow[s0 + koff];
    av.y = prow[s0 + koff + 1];
    const int r0 = s0 + koff, r1 = r0 + 1;
    const float* v0 = (r0 == pos) ? vnew : vbase + (size_t)r0 * HEAD_DIM;
    const float* v1 = (r1 == pos) ? vnew : vbase + (size_t)r1 * HEAD_DIM;
    bv.x = v0[d0 + nm];
    bv.y = v1[d0 + nm];
    c = __builtin_amdgcn_wmma_f32_16x16x4_f32(false, av, false, bv,
                                              (short)0, c, false, false);
  }
  if (lane < 16) {
#pragma unroll
    for (int r = 0; r < 8; ++r)
      attn[(size_t)(g * 8 + r) * HEAD_DIM + d0 + lane] = c[r];
  }
}

// ---------------- Kernel 6: sigmoid gate -----------------------------------
__global__ void k_gate(const float* __restrict__ attn,
                       const float* __restrict__ qg,
                       float* __restrict__ gated) {
  const int i = blockIdx.x * 256 + threadIdx.x;  // 0..8191
  const int h = i >> 8, d = i & 255;
  const float gt = qg[h * 512 + 256 + d];
  gated[i] = attn[i] * (1.f / (1.f + expf(-gt)));
}

// ---------------- Kernel 7: output GEMV (4096 x 8192) ----------------------
__global__ void k_gemv_o(const float* __restrict__ gated,
                         const float* __restrict__ ow,
                         float* __restrict__ out) {
  __shared__ float xs[NUM_HEADS * HEAD_DIM];     // 32KB
  const int tid = threadIdx.x;
  for (int i = tid; i < NUM_HEADS * HEAD_DIM; i += 256) xs[i] = gated[i];
  __syncthreads();

  const int lane = tid & 31;
  const int wv = tid >> 5;
  const int row = blockIdx.x * 8 + wv;           // 0..4095

  const float4* w4 = reinterpret_cast<const float4*>(ow + (size_t)row * 8192);
  const float4* x4 = reinterpret_cast<const float4*>(xs);
  float acc = 0.f;
#pragma unroll 8
  for (int it = 0; it < 8192 / 128; ++it) {      // 64 iters
    float4 wv4 = w4[it * 32 + lane];
    float4 xv4 = x4[it * 32 + lane];
    acc += wv4.x * xv4.x + wv4.y * xv4.y + wv4.z * xv4.z + wv4.w * xv4.w;
  }
  for (int off = 16; off > 0; off >>= 1) acc += __shfl_down(acc, off, 32);
  if (lane == 0) out[row] = acc;
}

// ---------------------------------------------------------------------------
extern "C" void kernel_launch(void* const* d_in, const int* in_sizes, int n_in,
                              void* d_out, int out_size, void* d_ws, size_t ws_size,
                              hipStream_t stream) {
  const float* x      = (const float*)d_in[0];
  const int*   posp   = (const int*)d_in[1];
  // d_in[2] mask, d_in[5] onehot: unused (equivalent handling via `pos`)
  const float* kcache = (const float*)d_in[3];
  const float* vcache = (const float*)d_in[4];
  const float* qw     = (const float*)d_in[6];
  const float* kw     = (const float*)d_in[7];
  const float* vw     = (const float*)d_in[8];
  const float* ow     = (const float*)d_in[9];
  const float* qnw    = (const float*)d_in[10];
  const float* knw    = (const float*)d_in[11];
  float* out = (float*)d_out;
  float* ws  = (float*)d_ws;

  // workspace layout (floats)
  float* qg    = ws;                 // 16384
  float* klin  = ws + 16384;         // 1024
  float* vlin  = ws + 17408;         // 1024
  float* q_ws  = ws + 18432;         // 4*16*256 = 16384 (zero-padded rows)
  float* krot  = ws + 34816;         // 1024
  float* sp    = ws + 35840;         // 4*16*2048 = 131072 (scores -> probs)
  float* attn  = ws + 166912;        // 8192
  float* gated = ws + 175104;        // 8192  (total 733KB)

  k_gemv_qkv<<<2304, 256, 0, stream>>>(x, qw, kw, vw, qg, klin, vlin);
  k_norm_rope<<<40, 256, 0, stream>>>(qg, klin, posp, qnw, knw, q_ws, krot);
  k_scores<<<128, 128, 0, stream>>>(q_ws, krot, kcache, posp, sp);
  k_softmax<<<64, 256, 0, stream>>>(sp, posp);
  k_pv<<<16, 128, 0, stream>>>(sp, vlin, vcache, posp, attn);
  k_gate<<<32, 256, 0, stream>>>(attn, qg, gated);
  k_gemv_o<<<512, 256, 0, stream>>>(gated, ow, out);
}